// VectorQuantizer_28398323761312
// MI455X (gfx1250) — compile-verified
//
#include <hip/hip_runtime.h>
#include <hip/hip_bf16.h>

typedef __attribute__((ext_vector_type(2))) float v2f;
typedef __attribute__((ext_vector_type(8))) float v8f;

#define NTOK   65536      // 64*32*32 tokens
#define KCODE  512        // codebook entries
#define DIM    64         // embedding dim
#define BETA   0.25f

// ---------------------------------------------------------------------------
// Fragment loader for V_WMMA_F32_16X16X4_F32.
// ISA 7.12.2 (32-bit A 16x4): lanes 0-15 hold M=lane, {VGPR0=K0,VGPR1=K1};
// lanes 16-31 hold M=lane-16, {VGPR0=K2,VGPR1=K3}. B (4x16) is symmetric with
// N in place of M (B[k][n] = rowmajor[n][k]).  Both reduce to: lane L reads
// row (L&15), columns 4*s + 2*(L>=16) + {0,1} of a row-major [rows, 64] array.
// ---------------------------------------------------------------------------
__device__ __forceinline__ v2f load_frag16x4(const float* __restrict__ base,
                                             int row0, int s, int lane) {
  const int r   = row0 + (lane & 15);
  const int col = 4 * s + ((lane >> 4) << 1);
  const float* q = base + (size_t)r * DIM + col;
  v2f v;
  v.x = q[0];
  v.y = q[1];
  return v;
}

// ---------------------------------------------------------------------------
// Kernel 0: per-token and per-code squared norms + zero loss accumulator.
// ---------------------------------------------------------------------------
__global__ void __launch_bounds__(256)
vq_norms(const float* __restrict__ x, const float* __restrict__ emb,
         float* __restrict__ xnorm, float* __restrict__ enorm,
         float* __restrict__ acc) {
  const int n = blockIdx.x * blockDim.x + threadIdx.x;
  if (n < NTOK) {
    const float4* p = (const float4*)(x + (size_t)n * DIM);
    float s = 0.f;
#pragma unroll
    for (int i = 0; i < DIM / 4; ++i) {
      float4 v = p[i];
      s += v.x * v.x + v.y * v.y + v.z * v.z + v.w * v.w;
    }
    xnorm[n] = s;
  }
  if (n < KCODE) {
    const float4* p = (const float4*)(emb + (size_t)n * DIM);
    float s = 0.f;
#pragma unroll
    for (int i = 0; i < DIM / 4; ++i) {
      float4 v = p[i];
      s += v.x * v.x + v.y * v.y + v.z * v.z + v.w * v.w;
    }
    enorm[n] = s;
  }
  if (n == 0) *acc = 0.f;
}

// ---------------------------------------------------------------------------
// Kernel 1: sq[k, n] = ||e_k||^2 + ||x_n||^2 - 2 * dot(e_k, x_n)
// One wave per 16(M) x 256(N) strip; A fragments (full K=64) preloaded.
// grid = (64, 32), block = 128 (4 waves). EXEC is all-1s everywhere (no
// divergence) as required for WMMA.
// ---------------------------------------------------------------------------
__global__ void __launch_bounds__(128)
vq_gemm(const float* __restrict__ emb, const float* __restrict__ x,
        const float* __restrict__ enorm, const float* __restrict__ xnorm,
        float* __restrict__ sq) {
  const int lane  = threadIdx.x & 31;
  const int wave  = threadIdx.x >> 5;
  const int m0    = blockIdx.y * 16;                 // code tile (0..511 step 16)
  const int strip = blockIdx.x * 4 + wave;           // 0..255
  const int nbase = strip * 256;                     // 256 tokens per wave

  // Preload A fragments: 16 k-steps of the 16x64 codebook slab (32 VGPRs).
  v2f a[16];
#pragma unroll
  for (int s = 0; s < 16; ++s) a[s] = load_frag16x4(emb, m0, s, lane);

  // Epilogue row constants: C VGPR r maps to M = m0 + r + 8*(lane>=16).
  const int moff = (lane >> 4) << 3;
  float en[8];
#pragma unroll
  for (int r = 0; r < 8; ++r) en[r] = enorm[m0 + r + moff];

  const int ncol = lane & 15;

  for (int t = 0; t < 16; ++t) {
    const int n0 = nbase + t * 16;
    v8f c = {};
#pragma unroll
    for (int s = 0; s < 16; ++s) {
      v2f b = load_frag16x4(x, n0, s, lane);
      // 8 args: (neg_a, A, neg_b, B, c_mod, C, reuse_a, reuse_b)
      c = __builtin_amdgcn_wmma_f32_16x16x4_f32(
          /*neg_a=*/false, a[s], /*neg_b=*/false, b,
          /*c_mod=*/(short)0, c, /*reuse_a=*/false, /*reuse_b=*/false);
    }
    const float xn = xnorm[n0 + ncol];
#pragma unroll
    for (int r = 0; r < 8; ++r) {
      const int m = m0 + r + moff;
      const float d = en[r] + xn - 2.0f * c[r];
      sq[(size_t)m * NTOK + (size_t)(n0 + ncol)] = d;
    }
  }
}

// ---------------------------------------------------------------------------
// Kernel 2: per-token argmin over K (first-min tie-break == argmax(-sq)),
// gather code, straight-through output, squared-error partial sums.
// Column scan of sq is fully coalesced across lanes (consecutive n, same k).
// ---------------------------------------------------------------------------
__global__ void __launch_bounds__(256)
vq_argmin(const float* __restrict__ sq, const float* __restrict__ x,
          const float* __restrict__ emb, float* __restrict__ enc,
          float* __restrict__ idxf, float* __restrict__ acc) {
  const int n = blockIdx.x * 256 + threadIdx.x;

  const float* col = sq + n;
  float best = 3.402823466e38f;
  int   bi   = 0;
#pragma unroll 8
  for (int k = 0; k < KCODE; ++k) {
    const float d = col[(size_t)k * NTOK];
    if (d < best) { best = d; bi = k; }   // strict < keeps first occurrence
  }
  idxf[n] = (float)bi;

  const float4* e  = (const float4*)(emb + (size_t)bi * DIM);
  const float4* xi = (const float4*)(x + (size_t)n * DIM);
  float4*       o  = (float4*)(enc + (size_t)n * DIM);
  float s = 0.f;
#pragma unroll
  for (int i = 0; i < DIM / 4; ++i) {
    const float4 ev = e[i];
    const float4 xv = xi[i];
    const float dx0 = ev.x - xv.x, dx1 = ev.y - xv.y;
    const float dx2 = ev.z - xv.z, dx3 = ev.w - xv.w;
    float4 st;                       // STE: x + (e - x), computed as written
    st.x = xv.x + dx0; st.y = xv.y + dx1;
    st.z = xv.z + dx2; st.w = xv.w + dx3;
    o[i] = st;
    s += dx0 * dx0 + dx1 * dx1 + dx2 * dx2 + dx3 * dx3;
  }

  __shared__ float red[256];
  red[threadIdx.x] = s;
  __syncthreads();
#pragma unroll
  for (int off = 128; off > 0; off >>= 1) {
    if (threadIdx.x < off) red[threadIdx.x] += red[threadIdx.x + off];
    __syncthreads();
  }
  if (threadIdx.x == 0) atomicAdd(acc, red[0]);
}

// ---------------------------------------------------------------------------
// Kernel 3: loss = (1 + beta) * sum_sq_err / (N * D)
// ---------------------------------------------------------------------------
__global__ void vq_loss(const float* __restrict__ acc, float* __restrict__ loss) {
  *loss = acc[0] * (1.0f + BETA) / (float)((size_t)NTOK * DIM);
}

// ---------------------------------------------------------------------------
extern "C" void kernel_launch(void* const* d_in, const int* in_sizes, int n_in,
                              void* d_out, int out_size, void* d_ws, size_t ws_size,
                              hipStream_t stream) {
  const float* x   = (const float*)d_in[0];   // inputs  [64,32,32,64]
  const float* emb = (const float*)d_in[1];   // embeddings [512,64]

  float* out  = (float*)d_out;
  float* enc  = out;                                   // 4,194,304
  float* idxf = out + (size_t)NTOK * DIM;              // 65,536
  float* sq   = idxf + NTOK;                           // 33,554,432
  float* loss = sq + (size_t)KCODE * NTOK;             // 1

  float* xnorm = (float*)d_ws;           // 65536 floats
  float* enorm = xnorm + NTOK;           // 512 floats
  float* acc   = enorm + KCODE;          // 1 float

  vq_norms<<<NTOK / 256, 256, 0, stream>>>(x, emb, xnorm, enorm, acc);
  vq_gemm<<<dim3(64, 32), 128, 0, stream>>>(emb, x, enorm, xnorm, sq);
  vq_argmin<<<NTOK / 256, 256, 0, stream>>>(sq, x, emb, enc, idxf, acc);
  vq_loss<<<1, 1, 0, stream>>>(acc, loss);
}